// DynamicMin2DPool_63333587746929
// MI455X (gfx1250) — compile-verified
//
#include <hip/hip_runtime.h>
#include <cstdint>

// Adaptive min 2D pool: [128, 512, 512] f32 -> [128, 64] f32 (8x8 grid, 64x64 regions).
// Memory-bound streaming kernel using gfx1250 async global->LDS copies
// (ASYNCcnt-tracked), double-buffered, with wave32 shuffle min reduction.
// IOFFSET immediates (applied to both LDS and global addresses per ISA 15.18)
// remove per-issue address arithmetic.

#define GRIDP 8
#define NROWS 512
#define NCOLS 512
#define ROWS_PER_BLOCK 64                     // one grid-row band per workgroup
#define TILE_ROWS 8                           // 8 x 512 f32 = 16 KB per tile
#define NTILES (ROWS_PER_BLOCK / TILE_ROWS)   // 8 tiles
#define TILE_FLOATS (TILE_ROWS * NCOLS)       // 4096
#define TILE_BYTES (TILE_FLOATS * 4)          // 16384

__global__ __launch_bounds__(256)
void minpool_async_kernel(const float* __restrict__ in, float* __restrict__ out) {
    __shared__ float smem[2][TILE_FLOATS];   // 2 x 16 KB double buffer
    __shared__ float red[16];                // cross-wave reduction slots

    const int t   = threadIdx.x;             // 0..255
    const int blk = blockIdx.x;              // 0..1023
    const int b   = blk >> 3;                // batch
    const int gr  = blk & 7;                 // grid row

    const float* src =
        in + ((size_t)b * NROWS + (size_t)gr * ROWS_PER_BLOCK) * NCOLS;

    // Per-thread base addresses (16 B per lane). LDS byte address = low 32 bits
    // of the flat LDS-aperture pointer (ISA 10.2).
    const uint64_t gbase = (uint64_t)(uintptr_t)src + (uint64_t)t * 16u;
    const uint32_t l0 = (uint32_t)(uintptr_t)(void*)&smem[0][0] + (uint32_t)t * 16u;
    const uint32_t l1 = (uint32_t)(uintptr_t)(void*)&smem[1][0] + (uint32_t)t * 16u;

    // Issue one 16 KB tile as 4 async copies (32 lanes x 16 B each, coalesced).
    // INST_OFFSET is added to both the LDS and global addresses, so one base
    // register pair + immediates covers the whole tile. Tracked by ASYNCcnt.
    auto issue_tile = [&](int tile) {
        const uint64_t g = gbase + (uint64_t)tile * TILE_BYTES;
        const uint32_t l = (tile & 1) ? l1 : l0;
        asm volatile(
            "global_load_async_to_lds_b128 %0, %1, off\n\t"
            "global_load_async_to_lds_b128 %0, %1, off offset:4096\n\t"
            "global_load_async_to_lds_b128 %0, %1, off offset:8192\n\t"
            "global_load_async_to_lds_b128 %0, %1, off offset:12288"
            :: "v"(l), "v"(g)
            : "memory");
    };

    issue_tile(0);

    const int c4   = t & 127;   // fixed float4 column 0..127 -> one grid col gc = c4>>4
    const int rsub = t >> 7;    // 0/1: which rows of the tile this thread scans
    float m = __builtin_inff();

    for (int tile = 0; tile < NTILES; ++tile) {
        if (tile < NTILES - 1) {
            issue_tile(tile + 1);
            // own wave has 8 async ops in flight; the first 4 (previous tile)
            // retire in order -> asynccnt <= 4 means tile `tile` is in LDS
            asm volatile("s_wait_asynccnt 4" ::: "memory");
        } else {
            asm volatile("s_wait_asynccnt 0" ::: "memory");
        }
        __syncthreads();   // publish LDS tile to all waves

        const float4* buf = (const float4*)&smem[tile & 1][0];
#pragma unroll
        for (int r = rsub; r < TILE_ROWS; r += 2) {
            const float4 v = buf[r * (NCOLS / 4) + c4];
            m = fminf(m, fminf(fminf(v.x, v.y), fminf(v.z, v.w)));
        }
        __syncthreads();   // tile consumed; its buffer is reused at tile+2
    }

    // Wave32 reduction: each 16-lane half of a wave shares one grid column.
#pragma unroll
    for (int s = 1; s < 16; s <<= 1)
        m = fminf(m, __shfl_xor(m, s, 32));

    const int wave = t >> 5;
    const int lane = t & 31;
    if (lane == 0 || lane == 16) {
        const int gc = ((wave << 1) + (lane >> 4)) & 7;   // grid column of this half-wave
        red[gc * 2 + (wave >> 2)] = m;                    // two contributions per gc
    }
    __syncthreads();

    if (t < GRIDP) {
        const float r = fminf(red[2 * t], red[2 * t + 1]);
        out[(size_t)b * (GRIDP * GRIDP) + gr * GRIDP + t] = r;
    }
}

extern "C" void kernel_launch(void* const* d_in, const int* in_sizes, int n_in,
                              void* d_out, int out_size, void* d_ws, size_t ws_size,
                              hipStream_t stream) {
    (void)in_sizes; (void)n_in; (void)d_ws; (void)ws_size; (void)out_size;
    const float* in  = (const float*)d_in[0];
    float*       out = (float*)d_out;
    // 128 batches x 8 grid rows = 1024 workgroups, 256 threads each.
    minpool_async_kernel<<<dim3(1024), dim3(256), 0, stream>>>(in, out);
}